// Sparsegen_29618094473533
// MI455X (gfx1250) — compile-verified
//
#include <hip/hip_runtime.h>

// Sparsegen (sigma=0 -> sparsemax) over rows of a 32768x2048 f32 matrix.
// Memory-bound (512 MB traffic -> ~22us @ 23.3 TB/s). One wave32 per row,
// row held in 64 VGPRs/lane. tau found by 30-step bisection on [max-1, max]
// + exact refinement, all via wave32 shuffle reductions (no barriers).
// CDNA5 path: per-wave double-buffered GLOBAL_LOAD_ASYNC_TO_LDS_B128 prefetch
// of the next row (ASYNCcnt), hiding HBM latency behind the bisection chain.

constexpr int kCols         = 2048;
constexpr int kLanes        = 32;               // wave32
constexpr int kPerLane      = kCols / kLanes;   // 64 floats per lane
constexpr int kChunks       = kPerLane / 4;     // 16 x float4 per lane
constexpr int kWavesPerBlk  = 4;                // 128 threads per block
constexpr int kBisectIters  = 30;

typedef int v4i __attribute__((ext_vector_type(4)));
typedef __attribute__((address_space(1))) v4i* gptr_v4i;
typedef __attribute__((address_space(3))) v4i* lptr_v4i;

#if defined(__HIP_DEVICE_COMPILE__) && \
    __has_builtin(__builtin_amdgcn_global_load_async_to_lds_b128) && \
    __has_builtin(__builtin_amdgcn_s_wait_asynccnt)
#define USE_ASYNC_LDS 1
#else
#define USE_ASYNC_LDS 0
#endif

__device__ __forceinline__ void stage_row_to_lds(const float* __restrict__ g,
                                                 float* __restrict__ lbuf,
                                                 int lane) {
#if USE_ASYNC_LDS
#pragma unroll
  for (int c = 0; c < kChunks; ++c) {
    const float* gp = g + c * 128 + lane * 4;
    float*       lp = lbuf + c * 128 + lane * 4;
    __builtin_amdgcn_global_load_async_to_lds_b128(
        (gptr_v4i)(void*)gp,
        (lptr_v4i)(void*)lp,
        /*offset=*/0, /*cpol=*/0);
  }
#else
  // Fallback: synchronous staging (same-wave DS ordering handled by compiler),
  // still exercises the gfx1250 prefetch path.
#pragma unroll
  for (int c = 0; c < kChunks; ++c) {
    __builtin_prefetch(g + c * 128 + lane * 4, 0, 1);
    *(float4*)(lbuf + c * 128 + lane * 4) =
        *(const float4*)(g + c * 128 + lane * 4);
  }
#endif
}

__device__ __forceinline__ void wait_staged() {
#if USE_ASYNC_LDS
  __builtin_amdgcn_s_wait_asynccnt(0);
#endif
  asm volatile("" ::: "memory");  // keep DS reads below the wait
}

__device__ __forceinline__ float waveReduceMax(float v) {
#pragma unroll
  for (int off = 16; off > 0; off >>= 1)
    v = fmaxf(v, __shfl_xor(v, off, kLanes));
  return v;
}

__device__ __forceinline__ float waveReduceSum(float v) {
#pragma unroll
  for (int off = 16; off > 0; off >>= 1)
    v += __shfl_xor(v, off, kLanes);
  return v;
}

__global__ __launch_bounds__(kWavesPerBlk * kLanes)
void sparsemax_rows_kernel(const float* __restrict__ x,
                           float* __restrict__ out,
                           int rows) {
  __shared__ __align__(16) float sbuf[kWavesPerBlk][2][kCols];

  const int lane      = threadIdx.x & (kLanes - 1);
  const int wave      = threadIdx.x >> 5;
  const int waveGid   = blockIdx.x * kWavesPerBlk + wave;
  const int waveCount = gridDim.x * kWavesPerBlk;

  int row = waveGid;
  if (row >= rows) return;

  // Prime the pipeline: stage first row into buffer 0.
  stage_row_to_lds(x + (long)row * kCols, &sbuf[wave][0][0], lane);

  int buf = 0;
  for (; row < rows; row += waveCount) {
    const int nextRow = row + waveCount;

    wait_staged();  // current row resident in LDS

    // LDS -> registers (ds_load_b128 x16), 64 floats per lane.
    float4 v[kChunks];
    {
      const float* lbuf = &sbuf[wave][buf][0];
#pragma unroll
      for (int c = 0; c < kChunks; ++c)
        v[c] = *(const float4*)(lbuf + c * 128 + lane * 4);
    }

    // Kick off async prefetch of the next row into the other buffer; it
    // overlaps with the ~30-iteration bisection below.
    if (nextRow < rows)
      stage_row_to_lds(x + (long)nextRow * kCols, &sbuf[wave][buf ^ 1][0], lane);
    buf ^= 1;

    // ---- row max ----
    float m = -3.402823466e38f;
#pragma unroll
    for (int c = 0; c < kChunks; ++c)
      m = fmaxf(m, fmaxf(fmaxf(v[c].x, v[c].y), fmaxf(v[c].z, v[c].w)));
    m = waveReduceMax(m);

    // ---- bisection for tau on [m-1, m]; f(t)=sum relu(x-t) is decreasing ----
    float lo = m - 1.0f;
    float hi = m;
#pragma unroll 1
    for (int it = 0; it < kBisectIters; ++it) {
      const float mid = 0.5f * (lo + hi);
      float s = 0.0f;
#pragma unroll
      for (int c = 0; c < kChunks; ++c) {
        s += fmaxf(v[c].x - mid, 0.0f);
        s += fmaxf(v[c].y - mid, 0.0f);
        s += fmaxf(v[c].z - mid, 0.0f);
        s += fmaxf(v[c].w - mid, 0.0f);
      }
      s = waveReduceSum(s);
      if (s > 1.0f) lo = mid; else hi = mid;   // wave-uniform branch
    }

    // ---- exact refinement: support = {x_i > thr}, tau = (sum - 1)/k ----
    const float thr = 0.5f * (lo + hi);
    float s = 0.0f, k = 0.0f;
#pragma unroll
    for (int c = 0; c < kChunks; ++c) {
      if (v[c].x > thr) { s += v[c].x; k += 1.0f; }
      if (v[c].y > thr) { s += v[c].y; k += 1.0f; }
      if (v[c].z > thr) { s += v[c].z; k += 1.0f; }
      if (v[c].w > thr) { s += v[c].w; k += 1.0f; }
    }
    s = waveReduceSum(s);
    k = waveReduceSum(k);
    const float tau = (s - 1.0f) / k;

    // ---- emit relu(x - tau), coalesced b128 stores ----
    float* orow = out + (long)row * kCols;
#pragma unroll
    for (int c = 0; c < kChunks; ++c) {
      float4 o;
      o.x = fmaxf(v[c].x - tau, 0.0f);
      o.y = fmaxf(v[c].y - tau, 0.0f);
      o.z = fmaxf(v[c].z - tau, 0.0f);
      o.w = fmaxf(v[c].w - tau, 0.0f);
      *(float4*)(orow + c * 128 + lane * 4) = o;
    }
  }
}

extern "C" void kernel_launch(void* const* d_in, const int* in_sizes, int n_in,
                              void* d_out, int out_size, void* d_ws, size_t ws_size,
                              hipStream_t stream) {
  const float* x   = (const float*)d_in[0];
  float*       out = (float*)d_out;

  const int total = in_sizes[0];
  const int rows  = total / kCols;   // 32768 for the reference shapes

  int blocks = 2048;                 // 8192 waves -> 4 rows/wave pipeline depth
  const int maxBlocks = (rows + kWavesPerBlk - 1) / kWavesPerBlk;
  if (blocks > maxBlocks) blocks = maxBlocks;
  if (blocks < 1) blocks = 1;

  sparsemax_rows_kernel<<<blocks, kWavesPerBlk * kLanes, 0, stream>>>(x, out, rows);
}